// SimpleGAT_69088843924163
// MI455X (gfx1250) — compile-verified
//
#include <hip/hip_runtime.h>
#include <hip/hip_bf16.h>

typedef __attribute__((ext_vector_type(2))) float v2f;
typedef __attribute__((ext_vector_type(8))) float v8f;

#define IN_CH   128
#define HCOUT   128   // HEADS * OUT_CH
#define HEADS   4
#define NEG_SLOPE 0.2f

// B-fragment-swizzled LDS row stride in floats: 128 cols * 2 K-values + 32 pad
// (stride % 256 == 32 dwords -> upper half-wave reads banks 32..63: conflict-free)
#define BROW 288

// ---------------------------------------------------------------------------
// Kernel 1: h = x @ W  via V_WMMA_F32_16X16X4_F32
// Block: 256 threads = 8 waves. 4 waves in M (64 rows) x 2 waves in N (128 cols).
// Each wave: 16x64 strip = 4 accumulators of 16x16, K=128 in steps of 4.
// W staged through LDS in two K-chunks of 64 rows, pre-swizzled so each
// B fragment is one aligned ds_load_b64. A fragments preloaded per K-half.
// ---------------------------------------------------------------------------
__global__ __launch_bounds__(256) void gat_gemm_wmma(
    const float* __restrict__ x, const float* __restrict__ W,
    float* __restrict__ h, int N)
{
  // 32 (ks,half) rows x 288 floats = 36,864 B
  __shared__ float sBf[32 * BROW];

  const int wid  = threadIdx.x >> 5;
  const int lane = threadIdx.x & 31;
  const int mW   = wid & 3;          // wave row group
  const int nW   = wid >> 2;         // wave col group (0..1)
  const int rowBase = blockIdx.x * 64 + mW * 16;
  const int colBase = nW * 64;

  // A-fragment addressing (ISA 7.12.2, 32-bit A 16x4):
  //   lanes 0-15 : M = lane,     K = k+0, k+1
  //   lanes 16-31: M = lane-16,  K = k+2, k+3
  const int aRow  = rowBase + (lane & 15);
  const int aRowC = aRow < N ? aRow : (N - 1);     // clamp (stores are guarded)
  const int half  = lane >> 4;                     // 0 or 1
  const int kOff  = half << 1;                     // 0 or 2
  const float* xrow = x + (size_t)aRowC * IN_CH + kOff;

  const int bCol = lane & 15;                      // B 4x16: N = lane&15

  v8f acc0 = {}, acc1 = {}, acc2 = {}, acc3 = {};

  const float4* Wv = (const float4*)W;

  for (int kp = 0; kp < 2; ++kp) {
    __syncthreads();
    // Stage 64x128 chunk of W, swizzled to B-fragment order:
    //   sBf[(rL>>1)*BROW + c*2 + (rL&1)] = W[r][c],  rL = r - kp*64
    for (int i = threadIdx.x; i < 2048; i += 256) {
      const int gi = kp * 2048 + i;     // float4 index into W
      float4 w4 = Wv[gi];
      const int rL = (gi >> 5) - kp * 64;
      const int c  = (gi & 31) << 2;
      float* dst = &sBf[(rL >> 1) * BROW + (rL & 1)];
      dst[(c + 0) * 2] = w4.x;
      dst[(c + 1) * 2] = w4.y;
      dst[(c + 2) * 2] = w4.z;
      dst[(c + 3) * 2] = w4.w;
    }
    __syncthreads();

    // Preload this K-half's A fragments: 16 x global_load_b64, one wait.
    v2f aReg[16];
    #pragma unroll
    for (int ks = 0; ks < 16; ++ks)
      aReg[ks] = *(const v2f*)(xrow + kp * 64 + ks * 4);

    #pragma unroll
    for (int ks = 0; ks < 16; ++ks) {
      const v2f* bp = (const v2f*)(sBf + ((ks << 1) + half) * BROW)
                      + colBase + bCol;
      v2f b0 = bp[0];
      v2f b1 = bp[16];
      v2f b2 = bp[32];
      v2f b3 = bp[48];
      acc0 = __builtin_amdgcn_wmma_f32_16x16x4_f32(false, aReg[ks], false, b0, (short)0, acc0, false, false);
      acc1 = __builtin_amdgcn_wmma_f32_16x16x4_f32(false, aReg[ks], false, b1, (short)0, acc1, false, false);
      acc2 = __builtin_amdgcn_wmma_f32_16x16x4_f32(false, aReg[ks], false, b2, (short)0, acc2, false, false);
      acc3 = __builtin_amdgcn_wmma_f32_16x16x4_f32(false, aReg[ks], false, b3, (short)0, acc3, false, false);
    }
  }

  // D layout: VGPR r -> row = r (+8 for lanes 16-31), col = lane&15
  const int rOff  = half << 3;   // 0 or 8
  const int cLane = lane & 15;
  #pragma unroll
  for (int r = 0; r < 8; ++r) {
    const int row = rowBase + r + rOff;
    if (row < N) {
      float* hp = h + (size_t)row * HCOUT + colBase + cLane;
      hp[0]  = acc0[r];
      hp[16] = acc1[r];
      hp[32] = acc2[r];
      hp[48] = acc3[r];
    }
  }
}

// ---------------------------------------------------------------------------
// Kernel 2: per-node attention dots. One wave per node (wave32).
// lane covers channels lane*4..lane*4+3; head = lane>>3 (32 ch = 8 lanes).
// h is L2-resident right after kernel 1, so this re-read is cheap.
// ---------------------------------------------------------------------------
__global__ __launch_bounds__(256) void gat_dots(
    const float* __restrict__ h, const float* __restrict__ att_src,
    const float* __restrict__ att_dst, float* __restrict__ a_src,
    float* __restrict__ a_dst, int N)
{
  const int node = (int)((blockIdx.x * (size_t)blockDim.x + threadIdx.x) >> 5);
  const int lane = threadIdx.x & 31;
  if (node >= N) return;

  float4 hv = ((const float4*)(h + (size_t)node * HCOUT))[lane];
  float4 as = ((const float4*)att_src)[lane];
  float4 ad = ((const float4*)att_dst)[lane];

  float ps = hv.x * as.x + hv.y * as.y + hv.z * as.z + hv.w * as.w;
  float pd = hv.x * ad.x + hv.y * ad.y + hv.z * ad.z + hv.w * ad.w;

  // reduce within 8-lane head groups
  #pragma unroll
  for (int off = 1; off < 8; off <<= 1) {
    ps += __shfl_xor(ps, off, 32);
    pd += __shfl_xor(pd, off, 32);
  }

  if ((lane & 7) == 0) {
    const int head = lane >> 3;
    a_src[(size_t)node * HEADS + head] = ps;
    a_dst[(size_t)node * HEADS + head] = pd;
  }
}

// ---------------------------------------------------------------------------
// Kernel 3: init m = -inf, denom = 0, out = bias (d_out is poisoned by harness)
// ---------------------------------------------------------------------------
__global__ __launch_bounds__(256) void gat_init(
    float* __restrict__ m, float* __restrict__ denom,
    float* __restrict__ out, const float* __restrict__ bias, int N)
{
  const size_t i = blockIdx.x * (size_t)blockDim.x + threadIdx.x;
  if (i < (size_t)N * HEADS) { m[i] = -3.0e38f; denom[i] = 0.0f; }
  if (i < (size_t)N * HCOUT) out[i] = bias[i & (HCOUT - 1)];
}

// ---------------------------------------------------------------------------
// Kernel 4: edge logits (leaky relu) + segment max via float atomicMax
// (lowers to global_atomic_max_num_f32). One thread per edge, all 4 heads.
// ---------------------------------------------------------------------------
__global__ __launch_bounds__(256) void gat_logits(
    const int* __restrict__ ei, const float* __restrict__ a_src,
    const float* __restrict__ a_dst, float* __restrict__ ebuf,
    float* __restrict__ m, int E, int N)
{
  const int idx = (int)(blockIdx.x * (size_t)blockDim.x + threadIdx.x);
  if (idx >= E + N) return;
  int s, d;
  if (idx < E) { s = ei[idx]; d = ei[E + idx]; }
  else         { s = d = idx - E; }

  #pragma unroll
  for (int hd = 0; hd < HEADS; ++hd) {
    float v = a_src[(size_t)s * HEADS + hd] + a_dst[(size_t)d * HEADS + hd];
    v = v > 0.0f ? v : NEG_SLOPE * v;
    ebuf[(size_t)idx * HEADS + hd] = v;
    atomicMax(&m[(size_t)d * HEADS + hd], v);
  }
}

// ---------------------------------------------------------------------------
// Kernel 5: expe = exp(e - m[dst]); denom[dst] += expe (stable softmax)
// ---------------------------------------------------------------------------
__global__ __launch_bounds__(256) void gat_expsum(
    const int* __restrict__ ei, float* __restrict__ ebuf,
    const float* __restrict__ m, float* __restrict__ denom, int E, int N)
{
  const int idx = (int)(blockIdx.x * (size_t)blockDim.x + threadIdx.x);
  if (idx >= E + N) return;
  const int d = (idx < E) ? ei[E + idx] : (idx - E);

  #pragma unroll
  for (int hd = 0; hd < HEADS; ++hd) {
    float v = __expf(ebuf[(size_t)idx * HEADS + hd] - m[(size_t)d * HEADS + hd]);
    ebuf[(size_t)idx * HEADS + hd] = v;
    atomicAdd(&denom[(size_t)d * HEADS + hd], v);
  }
}

// ---------------------------------------------------------------------------
// Kernel 6: scatter-aggregate. One wave per edge; each lane: float4 of h[src],
// scaled by alpha(head), 4x global_atomic_add_f32 into out[dst]. h/out are
// L2-resident (192MB L2 >> 51MB working set) so atomics resolve at L2.
// Prefetch h[src] row before the dependent alpha loads.
// ---------------------------------------------------------------------------
__global__ __launch_bounds__(256) void gat_aggregate(
    const int* __restrict__ ei, const float* __restrict__ h,
    const float* __restrict__ ebuf, const float* __restrict__ denom,
    float* __restrict__ out, int E, int N)
{
  const int edge = (int)((blockIdx.x * (size_t)blockDim.x + threadIdx.x) >> 5);
  const int lane = threadIdx.x & 31;
  if (edge >= E + N) return;
  int s, d;
  if (edge < E) { s = ei[edge]; d = ei[E + edge]; }
  else          { s = d = edge - E; }

  const float* hrow = h + (size_t)s * HCOUT;
  __builtin_prefetch(hrow + lane * 4, 0, 3);   // global_prefetch_b8

  const int head = lane >> 3;
  const float alpha = ebuf[(size_t)edge * HEADS + head] /
                      denom[(size_t)d * HEADS + head];

  float4 hv = ((const float4*)hrow)[lane];
  float* op = out + (size_t)d * HCOUT + lane * 4;
  atomicAdd(op + 0, hv.x * alpha);
  atomicAdd(op + 1, hv.y * alpha);
  atomicAdd(op + 2, hv.z * alpha);
  atomicAdd(op + 3, hv.w * alpha);
}

// ---------------------------------------------------------------------------
extern "C" void kernel_launch(void* const* d_in, const int* in_sizes, int n_in,
                              void* d_out, int out_size, void* d_ws, size_t ws_size,
                              hipStream_t stream)
{
  const float* x       = (const float*)d_in[0];
  const int*   ei      = (const int*)  d_in[1];
  const float* W       = (const float*)d_in[2];
  const float* att_src = (const float*)d_in[3];
  const float* att_dst = (const float*)d_in[4];
  const float* bias    = (const float*)d_in[5];

  const int N = in_sizes[0] / IN_CH;
  const int E = in_sizes[1] / 2;
  const int Etot = E + N;

  float* out = (float*)d_out;

  // workspace partition (floats)
  float* ws    = (float*)d_ws;
  float* h     = ws;                              // N*128
  float* a_src = h     + (size_t)N * HCOUT;       // N*4
  float* a_dst = a_src + (size_t)N * HEADS;       // N*4
  float* m     = a_dst + (size_t)N * HEADS;       // N*4
  float* denom = m     + (size_t)N * HEADS;       // N*4
  float* ebuf  = denom + (size_t)N * HEADS;       // (E+N)*4

  gat_gemm_wmma<<<(N + 63) / 64, 256, 0, stream>>>(x, W, h, N);
  gat_dots<<<(N + 7) / 8, 256, 0, stream>>>(h, att_src, att_dst, a_src, a_dst, N);
  gat_init<<<((size_t)N * HCOUT + 255) / 256, 256, 0, stream>>>(m, denom, out, bias, N);
  gat_logits<<<(Etot + 255) / 256, 256, 0, stream>>>(ei, a_src, a_dst, ebuf, m, E, N);
  gat_expsum<<<(Etot + 255) / 256, 256, 0, stream>>>(ei, ebuf, m, denom, E, N);
  gat_aggregate<<<(Etot + 7) / 8, 256, 0, stream>>>(ei, h, ebuf, denom, out, E, N);
}